// FrameBuilder_2456721293909
// MI455X (gfx1250) — compile-verified
//
#include <hip/hip_runtime.h>
#include <math.h>

// FrameBuilder: gather p0,p1,p2 per triplet, build frame matching the JAX ref:
//   z = (d10 + eps*Z)/(|d10|+eps); y = (z x d20 + eps*Y)/(|.|+eps);
//   x = (y x z + eps*X)/(|.|+eps); out = [p0, x, y, z]  (N,T,4,3) f32
//
// Memory-bound (~170 MB end-to-end => ~7.3us at 23.3 TB/s). NT cache hints
// keep the 24 MB points table resident in the 192 MB L2 while triplets (in)
// and frames (out) stream past it.

#define EPSF 1e-6f

// Native clang vector type: valid operand for __builtin_nontemporal_store,
// lowers to global_store_b128 with TH=NT.
typedef float v4f __attribute__((ext_vector_type(4)));

__global__ __launch_bounds__(256) void FrameBuilder_2456721293909_kernel(
    const float* __restrict__ points,    // (N, A, 3) f32
    const int*  __restrict__ triplets,   // (N, T, 3) int32 (harness integer mapping)
    float* __restrict__ out,             // (N, T, 4, 3) f32
    int A, int T)
{
    const int t = blockIdx.x * blockDim.x + threadIdx.x;
    if (t >= T) return;
    const int n = blockIdx.y;

    const long long e  = (long long)n * T + t;   // flat element id
    const long long tb = e * 3;

    // Streamed, never-reused index loads: non-temporal.
    int i0 = __builtin_nontemporal_load(triplets + tb + 0);
    int i1 = __builtin_nontemporal_load(triplets + tb + 1);
    int i2 = __builtin_nontemporal_load(triplets + tb + 2);

    const int Am1 = A - 1;
    i0 = i0 < 0 ? 0 : (i0 > Am1 ? Am1 : i0);
    i1 = i1 < 0 ? 0 : (i1 > Am1 ? Am1 : i1);
    i2 = i2 < 0 ? 0 : (i2 > Am1 ? Am1 : i2);

    // Random gather from the per-batch points slice (L2-resident, default RT hint).
    const float* __restrict__ p = points + (long long)n * A * 3;
    const int o0 = i0 * 3, o1 = i1 * 3, o2 = i2 * 3;
    const float p0x = p[o0], p0y = p[o0 + 1], p0z = p[o0 + 2];
    const float p1x = p[o1], p1y = p[o1 + 1], p1z = p[o1 + 2];
    const float p2x = p[o2], p2y = p[o2 + 1], p2z = p[o2 + 2];

    const float d10x = p1x - p0x, d10y = p1y - p0y, d10z = p1z - p0z;
    const float d20x = p2x - p0x, d20y = p2y - p0y, d20z = p2z - p0z;

    // zaxis = (d10 + eps*Z) / (|d10| + eps)
    const float nz = sqrtf(d10x * d10x + d10y * d10y + d10z * d10z);
    const float rz = 1.0f / (nz + EPSF);
    const float zx = d10x * rz;
    const float zy = d10y * rz;
    const float zz = (d10z + EPSF) * rz;

    // yaxis = (cross(z, d20) + eps*Y) / (|.| + eps)
    const float cx = zy * d20z - zz * d20y;
    const float cy = zz * d20x - zx * d20z;
    const float cz = zx * d20y - zy * d20x;
    const float ny = sqrtf(cx * cx + cy * cy + cz * cz);
    const float ry = 1.0f / (ny + EPSF);
    const float yx = cx * ry;
    const float yy = (cy + EPSF) * ry;
    const float yz = cz * ry;

    // xaxis = (cross(y, z) + eps*X) / (|.| + eps)
    const float ex = yy * zz - yz * zy;
    const float ey = yz * zx - yx * zz;
    const float ez = yx * zy - yy * zx;
    const float nx = sqrtf(ex * ex + ey * ey + ez * ez);
    const float rx = 1.0f / (nx + EPSF);
    const float xx = (ex + EPSF) * rx;
    const float xy = ey * rx;
    const float xz = ez * rx;

    // 48B per element, 16B-aligned: three b128 non-temporal stores.
    v4f* __restrict__ o = (v4f*)(out + e * 12);
    v4f r0 = {p0x, p0y, p0z, xx};
    v4f r1 = {xy,  xz,  yx,  yy};
    v4f r2 = {yz,  zx,  zy,  zz};
    __builtin_nontemporal_store(r0, o + 0);
    __builtin_nontemporal_store(r1, o + 1);
    __builtin_nontemporal_store(r2, o + 2);
}

extern "C" void kernel_launch(void* const* d_in, const int* in_sizes, int n_in,
                              void* d_out, int out_size, void* d_ws, size_t ws_size,
                              hipStream_t stream) {
    const float* points   = (const float*)d_in[0];  // (N, A, 3) f32
    const int*   triplets = (const int*)d_in[1];    // (N, T, 3) int
    float*       out      = (float*)d_out;          // (N, T, 4, 3) f32

    const int N = 64;                                   // per reference
    const int A = in_sizes[0] / (N * 3);                // 32768
    const long long total = (long long)in_sizes[1] / 3; // N*T
    const int T = (int)(total / N);                     // 32768

    dim3 block(256);
    dim3 grid((unsigned)((T + 255) / 256), (unsigned)N); // block never spans two n
    hipLaunchKernelGGL(FrameBuilder_2456721293909_kernel, grid, block, 0, stream,
                       points, triplets, out, A, T);
}